// GroupAssignment_6347961663872
// MI455X (gfx1250) — compile-verified
//
#include <hip/hip_runtime.h>
#include <hip/hip_bf16.h>

#define NN 256
#define TT 128
#define CC 256
#define HID 32
#define TH_F 1.0f
#define TAU_F 0.1f
#define EPS_BN_F 1e-5f
#define EPS_NORM_F 1e-8f

typedef __attribute__((ext_vector_type(2))) float v2f;
typedef __attribute__((ext_vector_type(8))) float v8f;

// x0 = v_rel[0,0,T-1,:] at flat offset (0*128+127)*256 = 32512
// x1 = v_rel[0,1,T-1,:] at flat offset (1*128+127)*256 = 65280
#define X0_OFF 32512
#define X1_OFF 65280

// ---------------------------------------------------------------------------
// K1: dm[n,m] = 0.5*(e(n,m)+e(m,n)), e = exp(w2 . BN(relu(w1 . d + b1)) + b2)
// BN folded into scale/shift; w2 folded into per-hidden weights.
// grid: 256 blocks (n), 256 threads (m)
// ---------------------------------------------------------------------------
__global__ void k_distmat(const float* __restrict__ v_rel,
                          const float* __restrict__ w1, const float* __restrict__ b1,
                          const float* __restrict__ bn_g, const float* __restrict__ bn_b,
                          const float* __restrict__ bn_m, const float* __restrict__ bn_v,
                          const float* __restrict__ w2, const float* __restrict__ b2,
                          float* __restrict__ dm) {
    __shared__ float s_w10[HID], s_w11[HID], s_b1[HID], s_ws2[HID], s_shift[HID];
    __shared__ float s_bias;
    int tid = threadIdx.x;
    if (tid < HID) {
        float scale = bn_g[tid] * rsqrtf(bn_v[tid] + EPS_BN_F);
        float shift = bn_b[tid] - bn_m[tid] * scale;
        float w2o   = w2[tid];
        s_w10[tid] = w1[tid * 2 + 0];
        s_w11[tid] = w1[tid * 2 + 1];
        s_b1[tid]  = b1[tid];
        s_ws2[tid] = w2o * scale;
        s_shift[tid] = w2o * shift;
    }
    __syncthreads();
    if (tid == 0) {
        float ss = b2[0];
        for (int o = 0; o < HID; ++o) ss += s_shift[o];
        s_bias = ss;
    }
    __syncthreads();

    int n = blockIdx.x, m = tid;
    float d0 = v_rel[X0_OFF + n] - v_rel[X0_OFF + m];
    float d1 = v_rel[X1_OFF + n] - v_rel[X1_OFF + m];

    float acc_f = s_bias, acc_r = s_bias;
    #pragma unroll
    for (int o = 0; o < HID; ++o) {
        float lin = s_w10[o] * d0 + s_w11[o] * d1;
        acc_f += s_ws2[o] * fmaxf( lin + s_b1[o], 0.0f);
        acc_r += s_ws2[o] * fmaxf(-lin + s_b1[o], 0.0f);
    }
    dm[n * NN + m] = 0.5f * (expf(acc_f) + expf(acc_r));
}

// ---------------------------------------------------------------------------
// K2: sig = sigmoid(-(dm-TH)/TAU); S[n,m] = sig[n,m] / (sum_n sig[n,m] + eps)
// grid: 256 blocks (column m), 256 threads (row n)
// ---------------------------------------------------------------------------
__global__ void k_signorm(const float* __restrict__ dm, float* __restrict__ S) {
    __shared__ float red[NN];
    int m = blockIdx.x, n = threadIdx.x;
    float d = dm[n * NN + m];
    float s = 1.0f / (1.0f + expf((d - TH_F) / TAU_F));
    red[n] = s;
    __syncthreads();
    for (int off = 128; off > 0; off >>= 1) {
        if (n < off) red[n] += red[n + off];
        __syncthreads();
    }
    float total = red[0];
    S[n * NN + m] = s / (total + EPS_NORM_F);
}

// ---------------------------------------------------------------------------
// K3: exact sequential label-merge scan (reference semantics) on ONE wave32.
// Labels live in LDS; wave lockstep means all lanes read L[r],L[c] before any
// lane writes. Then rank the surviving labels and emit group indices.
// grid: 1 block, 32 threads
// ---------------------------------------------------------------------------
__global__ void k_labels(const float* __restrict__ dm, float* __restrict__ grp_out) {
    volatile __shared__ int L[NN];
    volatile __shared__ int present[NN];
    volatile __shared__ int rnk[NN];
    int lane = threadIdx.x;  // 0..31

    #pragma unroll
    for (int k = 0; k < 8; ++k) L[lane + k * 32] = lane + k * 32;
    __syncthreads();

    for (int r = 1; r < NN; ++r) {
        for (int c = 0; c < r; ++c) {
            float d = dm[r * NN + c];
            if (d <= TH_F) {
                int lr = L[r];
                int lc = L[c];
                if (lr != lc) {
                    #pragma unroll
                    for (int k = 0; k < 8; ++k) {
                        int i = lane + k * 32;
                        if (L[i] == lr) L[i] = lc;
                    }
                }
            }
        }
    }
    __syncthreads();

    #pragma unroll
    for (int k = 0; k < 8; ++k) present[lane + k * 32] = 0;
    __syncthreads();
    #pragma unroll
    for (int k = 0; k < 8; ++k) present[L[lane + k * 32]] = 1;
    __syncthreads();
    for (int k = 0; k < 8; ++k) {
        int idx = lane + k * 32;
        int s = 0;
        for (int j = 0; j <= idx; ++j) s += present[j];
        rnk[idx] = s - 1;
    }
    __syncthreads();
    #pragma unroll
    for (int k = 0; k < 8; ++k) {
        int i = lane + k * 32;
        grp_out[i] = (float)rnk[L[i]];
    }
}

// ---------------------------------------------------------------------------
// K4: v_soft = V(32768x256) @ S(256x256) via v_wmma_f32_16x16x4_f32,
//     out = (v - v_soft) + v_soft   (reference numerics).
// One wave per 16x16 output tile; K chained in steps of 4 (64 WMMAs).
// grid: 4096 blocks x 256 threads (8 waves/block, 32768 tiles)
// ---------------------------------------------------------------------------
__global__ void k_group_gemm(const float* __restrict__ V, const float* __restrict__ S,
                             float* __restrict__ out) {
    int wave = threadIdx.x >> 5;
    int lane = threadIdx.x & 31;
    int tile = blockIdx.x * 8 + wave;     // 0..32767
    int row0 = (tile >> 4) * 16;          // M tile (2048 of them)
    int col0 = (tile & 15) * 16;          // N tile (16 of them)
    int l15  = lane & 15;
    int hi   = lane >> 4;                 // 0: K={0,1}, 1: K={2,3}

    const float* arow = V + (size_t)(row0 + l15) * NN;
    int col = col0 + l15;

    v8f c = {0.f, 0.f, 0.f, 0.f, 0.f, 0.f, 0.f, 0.f};
    for (int k = 0; k < NN; k += 4) {
        int kk = k + hi * 2;
        // A fragment: 16x4 tile, lane row = row0+l15, contiguous pair of K
        v2f a = *(const v2f*)(arow + kk);
        // B fragment: 4x16 tile of S, rows kk, kk+1 at column `col`
        v2f b;
        b.x = S[kk * NN + col];
        b.y = S[(kk + 1) * NN + col];
        c = __builtin_amdgcn_wmma_f32_16x16x4_f32(
                /*neg_a=*/false, a, /*neg_b=*/false, b,
                /*c_mod=*/(short)0, c, /*reuse_a=*/false, /*reuse_b=*/false);
    }

    // C/D layout: VGPR j -> row row0 + j + hi*8, col col0 + (lane&15)
    #pragma unroll
    for (int j = 0; j < 8; ++j) {
        size_t r = (size_t)(row0 + j + hi * 8);
        size_t idx = r * NN + (col0 + l15);
        float v  = V[idx];
        float vs = c[j];
        out[idx] = (v - vs) + vs;
    }
}

// ---------------------------------------------------------------------------
// K5: dist_mat_full = broadcast dm over T=128 -> (1,128,256,256), float4 stores
// grid: 8192 blocks x 256 threads (2,097,152 float4s)
// ---------------------------------------------------------------------------
__global__ void k_broadcast(const float4* __restrict__ dm4, float4* __restrict__ out4) {
    int idx = blockIdx.x * blockDim.x + threadIdx.x;
    int j = idx & 16383;   // 65536/4 float4 per t slice
    out4[idx] = dm4[j];
}

extern "C" void kernel_launch(void* const* d_in, const int* in_sizes, int n_in,
                              void* d_out, int out_size, void* d_ws, size_t ws_size,
                              hipStream_t stream) {
    const float* v_rel = (const float*)d_in[0];
    const float* w1    = (const float*)d_in[1];
    const float* b1    = (const float*)d_in[2];
    const float* bn_g  = (const float*)d_in[3];
    const float* bn_b  = (const float*)d_in[4];
    const float* bn_m  = (const float*)d_in[5];
    const float* bn_v  = (const float*)d_in[6];
    const float* w2    = (const float*)d_in[7];
    const float* b2    = (const float*)d_in[8];

    float* out = (float*)d_out;
    float* dm  = (float*)d_ws;           // 65536 floats
    float* S   = dm + NN * NN;           // 65536 floats

    float* v_grouped = out;                        // 8,388,608 floats
    float* grp       = out + 8388608;              //       256 floats
    float* distfull  = out + 8388864;              // 8,388,608 floats

    k_distmat  <<<256, 256, 0, stream>>>(v_rel, w1, b1, bn_g, bn_b, bn_m, bn_v, w2, b2, dm);
    k_signorm  <<<256, 256, 0, stream>>>(dm, S);
    k_labels   <<<1,    32, 0, stream>>>(dm, grp);
    k_group_gemm<<<4096, 256, 0, stream>>>(v_rel, S, v_grouped);
    k_broadcast<<<8192, 256, 0, stream>>>((const float4*)dm, (float4*)distfull);
}